// PolymerPhysicsPrior_9500467659170
// MI455X (gfx1250) — compile-verified
//
#include <hip/hip_runtime.h>
#include <hip/hip_bf16.h>
#include <math.h>

#define LBINS 401
#define BATCH 256
#define MIN_SEPB 2
#define NWS_FLOATS (LBINS + 3) // 401 bins + [ctcf_num, within_num, between_num]

typedef __attribute__((ext_vector_type(2))) float v2f;
typedef __attribute__((ext_vector_type(8))) float v8f;

// Wave32 sum-reduction via V_WMMA_F32_16X16X4_F32.
// A (16x4 f32, 2 VGPRs): vgpr0 = p  -> A[m,0]=p[m] (lanes 0-15), A[m,2]=p[m+16] (lanes 16-31)
//                        vgpr1 = 0  -> A[m,1]=A[m,3]=0
// B (4x16) = ones  =>  C[m,n] = p[m] + p[m+16]
// D layout: vgpr r: lanes0-15 = C[r,n], lanes16-31 = C[8+r,n]
// sum of 8 D vgprs per lane = half-sum; shfl_xor(16) completes it.
__device__ __forceinline__ float wave_reduce32(float p) {
  v2f a; a[0] = p;    a[1] = 0.0f;
  v2f b; b[0] = 1.0f; b[1] = 1.0f;
  v8f c = {};
  c = __builtin_amdgcn_wmma_f32_16x16x4_f32(false, a, false, b, (short)0, c,
                                            false, false);
  float v = c[0] + c[1] + c[2] + c[3] + c[4] + c[5] + c[6] + c[7];
  v += __shfl_xor(v, 16, 32);
  return v;
}

__device__ __forceinline__ float block_reduce256(float v, float* swave) {
  __syncthreads();                       // protect swave reuse
  float w = wave_reduce32(v);
  if ((threadIdx.x & 31) == 0) swave[threadIdx.x >> 5] = w;
  __syncthreads();
  return swave[0] + swave[1] + swave[2] + swave[3] +
         swave[4] + swave[5] + swave[6] + swave[7];
}

// Branchless per-element accumulation (lowered to v_cndmask + v_add, no exec branches).
__device__ __forceinline__ void accum_elem(float v, unsigned ci, unsigned cj, int d,
                                           float& t0, float& t1, float& t2) {
  bool anyp = (ci & cj & 4u) != 0u;
  bool conv = ((ci & 1u) != 0u) & ((cj & 2u) != 0u);
  t0 += (anyp & !conv) ? fmaxf(v, 0.0f) : 0.0f;
  bool far  = (d >= MIN_SEPB);
  bool same = (((ci ^ cj) & 8u) == 0u);
  t1 += (far &  same) ? v : 0.0f;
  t2 += (far & !same) ? v : 0.0f;
}

// ---------------- kernel 1: pack per-(b,l) codes, zero scratch ----------------
__global__ void k_codes(const float* __restrict__ logits,
                        const int* __restrict__ ctcf,
                        unsigned char* __restrict__ code,
                        float* __restrict__ wsf) {
  int idx = blockIdx.x * blockDim.x + threadIdx.x;
  if (idx < NWS_FLOATS) wsf[idx] = 0.0f;
  if (idx >= BATCH * LBINS) return;
  int o = ctcf[idx];
  unsigned c = 0;
  if (o == 1)  c |= 1u;                                   // forward
  if (o == -1) c |= 2u;                                   // reverse
  if (o != 0)  c |= 4u;                                   // any
  if (logits[2 * idx + 1] > logits[2 * idx + 0]) c |= 8u; // argmax (ties -> 0)
  code[idx] = (unsigned char)c;
}

// ---------------- kernel 2: single fused streaming pass ----------------
__global__ void __launch_bounds__(256) k_main(const float* __restrict__ cm,
                                              const unsigned char* __restrict__ code,
                                              float* __restrict__ gbins,
                                              float* __restrict__ gacc) {
  __shared__ float s_bins[LBINS];
  const int tid = threadIdx.x;
  for (int t = tid; t < LBINS; t += 256) s_bins[t] = 0.0f;
  __syncthreads();

  float t0 = 0.0f, t1 = 0.0f, t2 = 0.0f;  // hinge_num, within_num, between_num
  const int nrows = BATCH * LBINS;
  const int j1 = tid + 256;
  const bool has2 = (j1 < LBINS);         // waves 0-3 full, wave 4 partial

  for (int row = blockIdx.x; row < nrows; row += gridDim.x) {
    const int b = row / LBINS;
    const int i = row - b * LBINS;
    const unsigned ci = code[row];
    const unsigned char* __restrict__ cb = code + (size_t)b * LBINS;
    const size_t base = (size_t)row * LBINS;

    // Prefetch the next row this block will touch (gfx1250 global_prefetch_b8).
    const int nrow = row + gridDim.x;
    if (nrow < nrows) {
      const size_t nbase = (size_t)nrow * LBINS;
      __builtin_prefetch(cm + nbase + tid, 0, 0);
      __builtin_prefetch(cm + nbase + tid + 256, 0, 0);
    }

    // Peel both iterations; issue loads up front for MLP.
    float v0 = __builtin_nontemporal_load(cm + base + tid);
    float v1 = 0.0f;
    unsigned cj1 = 0;
    if (has2) {
      v1  = __builtin_nontemporal_load(cm + base + j1);
      cj1 = cb[j1];
    }
    unsigned cj0 = cb[tid];

    int d0 = (i > tid) ? (i - tid) : (tid - i);
    atomicAdd(&s_bins[d0], v0);                           // ds_add_f32
    accum_elem(v0, ci, cj0, d0, t0, t1, t2);

    if (has2) {
      int d1 = (i > j1) ? (i - j1) : (j1 - i);
      atomicAdd(&s_bins[d1], v1);
    }
    int d1c = (i > j1) ? (i - j1) : (j1 - i);
    accum_elem(v1, ci, cj1, d1c, t0, t1, t2);             // v1==0 when !has2
  }
  __syncthreads();

  // WMMA wave reductions (EXEC all-ones: uniform control flow here)
  float r0 = wave_reduce32(t0);
  float r1 = wave_reduce32(t1);
  float r2 = wave_reduce32(t2);
  if ((tid & 31) == 0) {
    atomicAdd(&gacc[0], r0);
    atomicAdd(&gacc[1], r1);
    atomicAdd(&gacc[2], r2);
  }
  for (int t = tid; t < LBINS; t += 256) atomicAdd(&gbins[t], s_bins[t]);
}

// ---------------- kernel 3: finalize (1 block, 256 threads) ----------------
__global__ void __launch_bounds__(256) k_final(const float* __restrict__ gbins,
                                               const float* __restrict__ gacc,
                                               const unsigned char* __restrict__ code,
                                               float* __restrict__ out) {
  __shared__ float swave[8];
  const int tid = threadIdx.x;

  // --- distance-decay regression, stage 1: n, sum(x), sum(y) ---
  float n = 0.0f, sx = 0.0f, sy = 0.0f;
  for (int d = tid; d < LBINS; d += 256) {
    float cnt = (d == 0) ? (float)LBINS : 2.0f * (float)(LBINS - d);
    float mean = gbins[d] / (cnt * (float)BATCH);
    bool valid = (d >= MIN_SEPB) && isfinite(mean) && (mean > 0.0f);
    if (valid) {
      float lx = logf(fmaxf((float)d, 1.0f));
      float ly = logf(mean + 1e-6f);
      n += 1.0f; sx += lx; sy += ly;
    }
  }
  n  = block_reduce256(n, swave);
  sx = block_reduce256(sx, swave);
  sy = block_reduce256(sy, swave);
  float nsafe = fmaxf(n, 1.0f);
  float xm = sx / nsafe, ym = sy / nsafe;

  // --- stage 2: centered sums ---
  float num = 0.0f, den = 0.0f;
  for (int d = tid; d < LBINS; d += 256) {
    float cnt = (d == 0) ? (float)LBINS : 2.0f * (float)(LBINS - d);
    float mean = gbins[d] / (cnt * (float)BATCH);
    bool valid = (d >= MIN_SEPB) && isfinite(mean) && (mean > 0.0f);
    if (valid) {
      float lx = logf(fmaxf((float)d, 1.0f));
      float ly = logf(mean + 1e-6f);
      num += (lx - xm) * (ly - ym);
      den += (lx - xm) * (lx - xm);
    }
  }
  num = block_reduce256(num, swave);
  den = block_reduce256(den, swave);
  float slope = num / (den + 1e-8f);
  float dist_loss = (n >= 5.0f) ? (slope + 0.85f) * (slope + 0.85f) : 0.0f;

  // --- analytic per-batch counts (thread == batch, BATCH==256) ---
  int A = 0, F = 0, R = 0, n1 = 0, adj = 0, prev = 0;
  const unsigned char* __restrict__ cb = code + (size_t)tid * LBINS;
  for (int l = 0; l < LBINS; ++l) {
    unsigned c = cb[l];
    A += (c >> 2) & 1u; F += c & 1u; R += (c >> 1) & 1u;
    int cp = (c >> 3) & 1u;
    n1 += cp;
    if (l > 0 && cp == prev) adj++;
    prev = cp;
  }
  float nc = (float)A * (float)A - (float)F * (float)R;            // A^2 - F*R
  float fn1 = (float)n1, fn0 = (float)(LBINS - n1);
  float same_cnt = fn0 * fn0 + fn1 * fn1 - (float)LBINS - 2.0f * (float)adj;
  const float mask_cnt = (float)LBINS * (float)LBINS - (float)LBINS
                         - 2.0f * (float)(LBINS - 1);
  float btw_cnt = mask_cnt - same_cnt;

  nc       = block_reduce256(nc, swave);
  same_cnt = block_reduce256(same_cnt, swave);
  btw_cnt  = block_reduce256(btw_cnt, swave);

  if (tid == 0) {
    float hinge = gacc[0] / (nc + 1e-6f);
    float ctcf_loss = (nc < 1.0f) ? 0.0f : hinge;
    float within  = gacc[1] / fmaxf(same_cnt, 1.0f);
    float between = gacc[2] / fmaxf(btw_cnt, 1.0f);
    float ratio = within / (fabsf(between) + 1e-6f);
    float comp_loss = fmaxf(1.5f - ratio, 0.0f);
    float total = 1.0f * dist_loss + 0.5f * ctcf_loss + 0.5f * comp_loss;
    out[0] = dist_loss; out[1] = ctcf_loss; out[2] = comp_loss; out[3] = total;
  }
}

extern "C" void kernel_launch(void* const* d_in, const int* in_sizes, int n_in,
                              void* d_out, int out_size, void* d_ws, size_t ws_size,
                              hipStream_t stream) {
  (void)in_sizes; (void)n_in; (void)out_size; (void)ws_size;
  const float* cm     = (const float*)d_in[0];            // (256,401,401) f32
  const float* logits = (const float*)d_in[1];            // (256,401,2)   f32
  const int*   ctcf   = (const int*)d_in[2];              // (256,401)     i32
  float* out = (float*)d_out;                             // 4 f32

  float* wsf = (float*)d_ws;
  float* g_bins = wsf;                                    // 401 floats
  float* g_acc  = wsf + LBINS;                            // 3 floats
  unsigned char* code = (unsigned char*)(wsf + NWS_FLOATS); // 256*401 bytes

  const int npts = BATCH * LBINS;
  k_codes<<<(npts + 255) / 256, 256, 0, stream>>>(logits, ctcf, code, wsf);
  k_main<<<2048, 256, 0, stream>>>(cm, code, g_bins, g_acc);
  k_final<<<1, 256, 0, stream>>>(g_bins, g_acc, code, out);
}